// KalmanFilterEstimator_90529320665747
// MI455X (gfx1250) — compile-verified
//
#include <hip/hip_runtime.h>

// Kalman filter scan for MI455X (gfx1250, wave32, WMMA).
// Phase 1: single workgroup runs the batch-independent Riccati/gain recursion
//          (exploits P symmetric), streams L_t to workspace. The 16x16 inverse
//          is a register-resident Gauss-Jordan in wave 0 using v_readlane_b32
//          broadcasts (no LDS pipe, no barriers) on the serial critical path.
// Phase 2: 16 workgroups (one per 16-row batch tile) replay the x recursion
//          with V_WMMA_F32_16X16X4_F32, reading L_{t-1} from L2. Streaming
//          inputs use non-temporal loads so Lbuf stays L2-resident.

typedef __attribute__((ext_vector_type(2))) float v2f;
typedef __attribute__((ext_vector_type(8))) float v8f;

#define TT  4096
#define BB  256
#define NXX 64
#define NYY 16
#define NUU 16
#define NDD 8

// Broadcast lane l's value to all lanes (l must be wave-uniform; here it is a
// compile-time constant from an unrolled loop) -> v_readlane_b32, VALU only.
__device__ __forceinline__ float bcast_lane(float x, int l) {
  return __int_as_float(__builtin_amdgcn_readlane(__float_as_int(x), l));
}

// 16x16 output tile accumulation: D += A(16xK) * B(Kx16), operands in LDS with
// arbitrary row/col strides (allows transposed reads without materialization).
// Fragment layouts per CDNA5 ISA 7.12.2 (f32, wave32):
//   A 16x4:  VGPR0 = (M=lane&15, K=kh), VGPR1 = K=kh+1, kh = (lane>>4)*2
//   B 4x16:  VGPR0 = (K=kh, N=lane&15), VGPR1 = K=kh+1
//   C/D 16x16: VGPR i = (M = i + (lane>>4)*8, N = lane&15)
__device__ __forceinline__ v8f wmma_mm(const float* A, int ras, int cas,
                                       const float* B, int rbs, int cbs,
                                       int K, v8f acc, int lane) {
  const int mn = lane & 15;
  const int kh = (lane >> 4) << 1;
#pragma unroll
  for (int k = 0; k < K; k += 4) {
    v2f a, b;
    a.x = A[mn * ras + (k + kh) * cas];
    a.y = A[mn * ras + (k + kh + 1) * cas];
    b.x = B[(k + kh) * rbs + mn * cbs];
    b.y = B[(k + kh + 1) * rbs + mn * cbs];
    acc = __builtin_amdgcn_wmma_f32_16x16x4_f32(false, a, false, b,
                                                (short)0, acc, false, false);
  }
  return acc;
}

__device__ __forceinline__ void store_tile(float* D, int ldd, v8f acc, int lane) {
  const int col = lane & 15;
  const int rb  = (lane >> 4) << 3;
#pragma unroll
  for (int i = 0; i < 8; ++i) D[(rb + i) * ldd + col] = acc[i];
}

// ---------------------------------------------------------------------------
// Phase 1: gain precompute. One workgroup, 8 waves. P kept symmetric:
//   T1 = Wfx P ;  P' = Wfx T1^T + I          (P Wfx^T == (Wfx P)^T)
//   PWfy = P' Wfy ;  S = R + Wfy^T PWfy ;  Sinv via register Gauss-Jordan
//   L = PWfy Sinv  -> Lbuf[t+1]
//   P'' = I - L PWfy^T                        (Wfy^T P' == (P' Wfy)^T)
// ---------------------------------------------------------------------------
__global__ __launch_bounds__(256) void kf_gains_kernel(
    const float* __restrict__ Wfx_g, const float* __restrict__ Wfy_g,
    float* __restrict__ Lbuf) {
  __shared__ float sWfx[NXX * NXX];   // 64x64 row-major
  __shared__ float sWfy[NXX * NYY];   // 64x16 row-major
  __shared__ float sP[NXX * NXX];
  __shared__ float sT1[NXX * NXX];
  __shared__ float sPWfy[NXX * NYY];
  __shared__ float sS[NYY * NYY];     // S = R + Wfy^T P Wfy
  __shared__ float sSinv[NYY * NYY];
  __shared__ float sL[NXX * NYY];

  const int tid  = threadIdx.x;
  const int lane = tid & 31;
  const int wave = tid >> 5;  // 8 waves

  for (int e = tid; e < NXX * NXX; e += 256) sWfx[e] = Wfx_g[e];
  for (int e = tid; e < NXX * NYY; e += 256) sWfy[e] = Wfy_g[e];
  for (int e = tid; e < NXX * NXX; e += 256) {
    int r = e >> 6, c = e & 63;
    sP[e] = (r == c) ? 1.0f : 0.0f;   // P0 = I
  }
  for (int e = tid; e < NXX * NYY; e += 256) Lbuf[e] = 0.0f;  // L_0 = 0
  __syncthreads();

  for (int t = 0; t < TT; ++t) {
    // Stage 1: T1 = Wfx @ P  (16 tiles over 8 waves)
    for (int tile = wave; tile < 16; tile += 8) {
      int mt = tile >> 2, nt = tile & 3;
      v8f acc = {};
      acc = wmma_mm(sWfx + mt * 16 * NXX, NXX, 1, sP + nt * 16, NXX, 1, NXX, acc, lane);
      store_tile(sT1 + mt * 16 * NXX + nt * 16, NXX, acc, lane);
    }
    __syncthreads();
    // Stage 2: P = Wfx @ T1^T + I   (B read transposed: B[k][n] = T1[n][k])
    for (int tile = wave; tile < 16; tile += 8) {
      int mt = tile >> 2, nt = tile & 3;
      v8f acc = {};
      acc = wmma_mm(sWfx + mt * 16 * NXX, NXX, 1, sT1 + nt * 16 * NXX, 1, NXX, NXX, acc, lane);
      int col = lane & 15, rb = (lane >> 4) << 3;
#pragma unroll
      for (int i = 0; i < 8; ++i) {
        int gr = mt * 16 + rb + i, gc = nt * 16 + col;
        sP[gr * NXX + gc] = acc[i] + ((gr == gc) ? 1.0f : 0.0f);
      }
    }
    __syncthreads();
    // Stage 3: PWfy = P @ Wfy  (4 tiles)
    if (wave < 4) {
      v8f acc = {};
      acc = wmma_mm(sP + wave * 16 * NXX, NXX, 1, sWfy, NYY, 1, NXX, acc, lane);
      store_tile(sPWfy + wave * 16 * NYY, NYY, acc, lane);
    }
    __syncthreads();
    // Stage 4+5: wave 0 only, no internal barriers (wave-lockstep + in-order
    // LDS within a wave give all needed ordering).
    if (wave == 0) {
      // S = R + Wfy^T @ PWfy  (A read transposed)
      v8f acc = {};
      acc = wmma_mm(sWfy, 1, NYY, sPWfy, NYY, 1, NXX, acc, lane);
      int col = lane & 15, rb = (lane >> 4) << 3;
#pragma unroll
      for (int i = 0; i < 8; ++i) {
        int r = rb + i;
        sS[r * NYY + col] = acc[i] + ((r == col) ? 1.0f : 0.0f);
      }
      // Register Gauss-Jordan on [S | I]: lane c owns augmented column c.
      // Pivot/factor broadcasts via v_readlane_b32 (constant source lanes);
      // S is SPD with diag >= 1, so no pivoting is required.
      float colv[16];
#pragma unroll
      for (int i = 0; i < 16; ++i)
        colv[i] = (lane < 16) ? sS[i * NYY + lane]
                              : ((i == (lane - 16)) ? 1.0f : 0.0f);
#pragma unroll
      for (int k = 0; k < 16; ++k) {
        float piv  = bcast_lane(colv[k], k);
        float rinv = 1.0f / piv;
        colv[k] *= rinv;
#pragma unroll
        for (int i = 0; i < 16; ++i) {
          if (i != k) {
            float f = bcast_lane(colv[i], k);
            colv[i] -= f * colv[k];
          }
        }
      }
      if (lane >= 16) {
#pragma unroll
        for (int i = 0; i < 16; ++i) sSinv[i * NYY + (lane - 16)] = colv[i];
      }
    }
    __syncthreads();
    // Stage 6: L = PWfy @ Sinv  -> LDS + Lbuf slot t+1
    if (wave < 4) {
      v8f acc = {};
      acc = wmma_mm(sPWfy + wave * 16 * NYY, NYY, 1, sSinv, NYY, 1, NYY, acc, lane);
      int col = lane & 15, rb = (lane >> 4) << 3;
      float* gL = Lbuf + (size_t)(t + 1) * (NXX * NYY);
#pragma unroll
      for (int i = 0; i < 8; ++i) {
        int gr = wave * 16 + rb + i;
        sL[gr * NYY + col] = acc[i];
        gL[gr * NYY + col] = acc[i];
      }
    }
    __syncthreads();
    // Stage 7: P = I - L @ PWfy^T  (B read transposed)
    for (int tile = wave; tile < 16; tile += 8) {
      int mt = tile >> 2, nt = tile & 3;
      v8f acc = {};
      acc = wmma_mm(sL + mt * 16 * NYY, NYY, 1, sPWfy + nt * 16 * NYY, 1, NYY, NYY, acc, lane);
      int col = lane & 15, rb = (lane >> 4) << 3;
#pragma unroll
      for (int i = 0; i < 8; ++i) {
        int gr = mt * 16 + rb + i, gc = nt * 16 + col;
        sP[gr * NXX + gc] = ((gr == gc) ? 1.0f : 0.0f) - acc[i];
      }
    }
    __syncthreads();
  }
}

// ---------------------------------------------------------------------------
// Phase 2: state scan. 16 workgroups x 128 threads (4 waves). Workgroup b
// owns batch rows [16b, 16b+16). Wave w owns column tile w of the 16x64 state.
// ---------------------------------------------------------------------------
__global__ __launch_bounds__(128) void kf_scan_kernel(
    const float* __restrict__ Ym, const float* __restrict__ U,
    const float* __restrict__ Dg,
    const float* __restrict__ Wfx_g, const float* __restrict__ bfx,
    const float* __restrict__ Wfu_g, const float* __restrict__ bfu,
    const float* __restrict__ Wfd_g, const float* __restrict__ bfd,
    const float* __restrict__ Wfy_g, const float* __restrict__ bfy_g,
    const float* __restrict__ Lbuf, float* __restrict__ Out) {
  __shared__ float sWfx[NXX * NXX];
  __shared__ float sWfu[NUU * NXX];
  __shared__ float sWfd[NDD * NXX];
  __shared__ float sWfy[NXX * NYY];
  __shared__ float sBsum[NXX];
  __shared__ float sBfy[NYY];
  __shared__ float sX[16 * NXX];
  __shared__ float sXp[16 * NXX];
  __shared__ float sL[NXX * NYY];
  __shared__ float sYm[16 * NYY];
  __shared__ float sU[16 * NUU];
  __shared__ float sD[16 * NDD];
  __shared__ float sInn[16 * NYY];

  const int tid  = threadIdx.x;
  const int lane = tid & 31;
  const int wave = tid >> 5;  // 4 waves
  const int r0   = blockIdx.x * 16;

  for (int e = tid; e < NXX * NXX; e += 128) sWfx[e] = Wfx_g[e];
  for (int e = tid; e < NUU * NXX; e += 128) sWfu[e] = Wfu_g[e];
  for (int e = tid; e < NDD * NXX; e += 128) sWfd[e] = Wfd_g[e];
  for (int e = tid; e < NXX * NYY; e += 128) sWfy[e] = Wfy_g[e];
  if (tid < NXX) sBsum[tid] = bfx[tid] + bfu[tid] + bfd[tid];
  if (tid < NYY) sBfy[tid] = bfy_g[tid];
  for (int e = tid; e < 16 * NXX; e += 128) sX[e] = 0.0f;  // x0 = 0
  __syncthreads();

  for (int t = 0; t < TT; ++t) {
    // Per-step operands (L_{t-1} = Lbuf[t]) + prefetch of step t+1.
    // Ym/U/D are streamed exactly once -> non-temporal loads keep them from
    // evicting the 16.8 MB Lbuf gain table (read 16x) out of L2.
    {
      const float* gL  = Lbuf + (size_t)t * (NXX * NYY);
      const float* gYm = Ym + (size_t)t * BB * NYY + r0 * NYY;
      const float* gU  = U  + (size_t)t * BB * NUU + r0 * NUU;
      const float* gD  = Dg + (size_t)t * BB * NDD + r0 * NDD;
      for (int e = tid; e < NXX * NYY; e += 128) sL[e] = gL[e];
      for (int e = tid; e < 16 * NYY; e += 128)
        sYm[e] = __builtin_nontemporal_load(gYm + e);
      for (int e = tid; e < 16 * NUU; e += 128)
        sU[e] = __builtin_nontemporal_load(gU + e);
      if (tid < 16 * NDD) sD[tid] = __builtin_nontemporal_load(gD + tid);
      if (t + 1 < TT) {
        const char* pL  = (const char*)(Lbuf + (size_t)(t + 1) * (NXX * NYY));
        const char* pYm = (const char*)(Ym + (size_t)(t + 1) * BB * NYY + r0 * NYY);
        const char* pU  = (const char*)(U  + (size_t)(t + 1) * BB * NUU + r0 * NUU);
        const char* pD  = (const char*)(Dg + (size_t)(t + 1) * BB * NDD + r0 * NDD);
        if (tid < 32) __builtin_prefetch(pL + tid * 128, 0, 0);
        if (tid < 8)  __builtin_prefetch(pYm + tid * 128, 0, 0);
        if (tid < 8)  __builtin_prefetch(pU + tid * 128, 0, 0);
        if (tid < 4)  __builtin_prefetch(pD + tid * 128, 0, 0);
      }
    }
    __syncthreads();
    // Stage 1: xp = x@Wfx + u@Wfu + d@Wfd + (bfx+bfu+bfd)
    {
      const int nt = wave;
      v8f acc = {};
      acc = wmma_mm(sX, NXX, 1, sWfx + nt * 16, NXX, 1, NXX, acc, lane);
      acc = wmma_mm(sU, NUU, 1, sWfu + nt * 16, NXX, 1, NUU, acc, lane);
      acc = wmma_mm(sD, NDD, 1, sWfd + nt * 16, NXX, 1, NDD, acc, lane);
      int col = lane & 15, rb = (lane >> 4) << 3;
      float bias = sBsum[nt * 16 + col];
#pragma unroll
      for (int i = 0; i < 8; ++i)
        sXp[(rb + i) * NXX + nt * 16 + col] = acc[i] + bias;
    }
    __syncthreads();
    // Stage 2: innov = ym - (xp@Wfy + bfy)   (one 16x16 tile, wave 0)
    if (wave == 0) {
      v8f acc = {};
      acc = wmma_mm(sXp, NXX, 1, sWfy, NYY, 1, NXX, acc, lane);
      int col = lane & 15, rb = (lane >> 4) << 3;
      float by = sBfy[col];
#pragma unroll
      for (int i = 0; i < 8; ++i) {
        int r = rb + i;
        sInn[r * NYY + col] = sYm[r * NYY + col] - (acc[i] + by);
      }
    }
    __syncthreads();
    // Stage 3: x = xp + innov @ L^T   (B read transposed: B[k][n] = L[n][k])
    {
      const int nt = wave;
      v8f acc = {};
      acc = wmma_mm(sInn, NYY, 1, sL + nt * 16 * NYY, 1, NYY, NYY, acc, lane);
      int col = lane & 15, rb = (lane >> 4) << 3;
#pragma unroll
      for (int i = 0; i < 8; ++i) {
        int idx = (rb + i) * NXX + nt * 16 + col;
        sX[idx] = sXp[idx] + acc[i];
      }
    }
    __syncthreads();
  }

  for (int e = tid; e < 16 * NXX; e += 128) {
    int row = e >> 6, col = e & 63;
    Out[(size_t)(r0 + row) * NXX + col] = sX[e];
  }
}

extern "C" void kernel_launch(void* const* d_in, const int* in_sizes, int n_in,
                              void* d_out, int out_size, void* d_ws, size_t ws_size,
                              hipStream_t stream) {
  const float* Ym  = (const float*)d_in[0];
  const float* U   = (const float*)d_in[1];
  const float* Dg  = (const float*)d_in[2];
  const float* Wfx = (const float*)d_in[3];
  const float* bfx = (const float*)d_in[4];
  const float* bfu = (const float*)d_in[6];
  const float* Wfu = (const float*)d_in[5];
  const float* Wfd = (const float*)d_in[7];
  const float* bfd = (const float*)d_in[8];
  const float* Wfy = (const float*)d_in[9];
  const float* bfy = (const float*)d_in[10];

  float* Lbuf = (float*)d_ws;  // (T+1) * 64 * 16 floats = 16.8 MB
  float* Out  = (float*)d_out;

  kf_gains_kernel<<<1, 256, 0, stream>>>(Wfx, Wfy, Lbuf);
  kf_scan_kernel<<<16, 128, 0, stream>>>(Ym, U, Dg, Wfx, bfx, Wfu, bfu,
                                         Wfd, bfd, Wfy, bfy, Lbuf, Out);
}